// GraphSageNNV1_28913719837490
// MI455X (gfx1250) — compile-verified
//
#include <hip/hip_runtime.h>
#include <math.h>

typedef __attribute__((ext_vector_type(2))) float v2f;
typedef __attribute__((ext_vector_type(8))) float v8f;

#define NNODES 50000
#define NEDGES 800000

// ---------------- zero fill ----------------
__global__ void gs_zero_kernel(float* p, long long n) {
  long long i = (long long)blockIdx.x * blockDim.x + threadIdx.x;
  if (i < n) p[i] = 0.0f;
}

// ---------------- degree ----------------
__global__ void gs_degree_kernel(const int* __restrict__ edst,
                                 float* __restrict__ deg, int E) {
  int e = blockIdx.x * blockDim.x + threadIdx.x;
  if (e < E) atomicAdd(&deg[edst[e]], 1.0f);
}

// ------------- edge scatter-add (mean-agg numerator) -------------
// One wave (32 lanes) per edge; lane-strided features -> coalesced.
__global__ __launch_bounds__(256)
void gs_scatter_kernel(const float* __restrict__ feat, int ldf,
                       const int* __restrict__ esrc,
                       const int* __restrict__ edst,
                       float* __restrict__ out, int ldo, int coff,
                       int E, int D) {
  int wave = (int)((blockIdx.x * blockDim.x + threadIdx.x) >> 5);
  int lane = threadIdx.x & 31;
  if (wave >= E) return;
  int s = esrc[wave];
  int d = edst[wave];
  const float* fs = feat + (long long)s * ldf;
  float* od = out + (long long)d * ldo + coff;
  for (int c = lane; c < D; c += 32)
    atomicAdd(&od[c], fs[c]);
}

// ------- finalize concat: H[:,0:D]=x, H[:,D:2D] *= 1/max(deg,1) -------
__global__ void gs_finalize_kernel(const float* __restrict__ x,
                                   const float* __restrict__ deg,
                                   float* __restrict__ H,
                                   int D, int ld, long long total) {
  long long i = (long long)blockIdx.x * blockDim.x + threadIdx.x;
  if (i >= total) return;
  long long n = i / D;
  int dcol = (int)(i % D);
  float inv = 1.0f / fmaxf(deg[n], 1.0f);
  H[n * ld + dcol] = x[n * (long long)D + dcol];
  H[n * ld + D + dcol] *= inv;
}

// ---- fp32 WMMA GEMM with LDS-staged weights: C = A[MxK]*B[KxN] + bias ----
// Block = 8 waves; block job = 8 m-tiles (128 rows) x 64 columns.
// B[:, cbase:cbase+64] staged transposed in LDS (pad 4 -> conflict-free b64
// reads, 8B aligned). Wave = one 16-row tile x 64 cols = 4 accumulators;
// per K-step of 4: 1 global_load_b64 (A) + 4 ds_load_b64 (B) + 4 WMMAs.
//
// V_WMMA_F32_16X16X4_F32 fragments (ISA 7.12.2, 32-bit):
//   A 16x4 : lane&15 = M, lane>>4 -> K+2, VGPR0/1 = K,K+1 (one v2f load)
//   B 4x16 : lane&15 = N, same K striping
//   C/D    : VGPR v -> M = v + 8*(lane>>4), N = lane&15
__global__ __launch_bounds__(256)
void gs_gemm_lds_kernel(const float* __restrict__ A,
                        const float* __restrict__ B,
                        const float* __restrict__ bias,
                        float* __restrict__ C,
                        int M, int K, int N, int do_relu) {
  extern __shared__ float ldsb[];            // [64][K+4]
  const int stride = K + 4;
  const int cgroups = N >> 6;
  const int mb = blockIdx.x / cgroups;
  const int cg = blockIdx.x - mb * cgroups;
  const int cbase = cg << 6;

  // cooperative stage: ldsb[c][k] = B[k][cbase+c]  (coalesced global reads,
  // LDS store bank = (4c + k) % 64 across consecutive threads -> ok)
  for (int idx = threadIdx.x; idx < (K << 6); idx += 256) {
    int k = idx >> 6;
    int c = idx & 63;
    ldsb[c * stride + k] = B[(long long)k * N + cbase + c];
  }
  __syncthreads();

  const int wave = threadIdx.x >> 5;
  const int lane = threadIdx.x & 31;
  const int l15  = lane & 15;
  const int kh   = lane >> 4;                // 0/1 -> K offset 0/2
  const int mtile = mb * 8 + wave;
  const int mtiles = M >> 4;

  if (mtile < mtiles) {                      // wave-uniform guard
    const int row = (mtile << 4) + l15;
    const float* arow = A + (long long)row * K + 2 * kh;   // 8B aligned
    const float* bp0 = &ldsb[(l15 +  0) * stride + 2 * kh];
    const float* bp1 = &ldsb[(l15 + 16) * stride + 2 * kh];
    const float* bp2 = &ldsb[(l15 + 32) * stride + 2 * kh];
    const float* bp3 = &ldsb[(l15 + 48) * stride + 2 * kh];

    v8f acc0 = {0.f,0.f,0.f,0.f,0.f,0.f,0.f,0.f};
    v8f acc1 = acc0, acc2 = acc0, acc3 = acc0;

    #pragma unroll 4
    for (int k0 = 0; k0 < K; k0 += 4) {
      v2f a  = *(const v2f*)(arow + k0);
      v2f b0 = *(const v2f*)(bp0 + k0);
      v2f b1 = *(const v2f*)(bp1 + k0);
      v2f b2 = *(const v2f*)(bp2 + k0);
      v2f b3 = *(const v2f*)(bp3 + k0);
      acc0 = __builtin_amdgcn_wmma_f32_16x16x4_f32(false, a, false, b0,
                                                   (short)0, acc0, false, false);
      acc1 = __builtin_amdgcn_wmma_f32_16x16x4_f32(false, a, false, b1,
                                                   (short)0, acc1, false, false);
      acc2 = __builtin_amdgcn_wmma_f32_16x16x4_f32(false, a, false, b2,
                                                   (short)0, acc2, false, false);
      acc3 = __builtin_amdgcn_wmma_f32_16x16x4_f32(false, a, false, b3,
                                                   (short)0, acc3, false, false);
    }

    const int mbase = (mtile << 4) + (kh << 3);  // lanes 16-31 own rows v+8
    v8f accs[4] = {acc0, acc1, acc2, acc3};
    #pragma unroll
    for (int t = 0; t < 4; ++t) {
      int col = cbase + (t << 4) + l15;
      float bs = bias[col];
      #pragma unroll
      for (int v = 0; v < 8; ++v) {
        float r = accs[t][v] + bs;
        if (do_relu) r = fmaxf(r, 0.0f);
        C[(long long)(mbase + v) * N + col] = r;
      }
    }
  }
}

// ---------------- row-wise log-softmax, wave per row, D=128 ----------------
__global__ __launch_bounds__(256)
void gs_logsoftmax_kernel(float* __restrict__ out, int M) {
  int wave = (int)((blockIdx.x * blockDim.x + threadIdx.x) >> 5);
  int lane = threadIdx.x & 31;
  if (wave >= M) return;
  float* row = out + (long long)wave * 128;
  float vals[4];
  float mx = -INFINITY;
  #pragma unroll
  for (int j = 0; j < 4; ++j) {
    vals[j] = row[lane + 32 * j];
    mx = fmaxf(mx, vals[j]);
  }
  #pragma unroll
  for (int off = 16; off > 0; off >>= 1)
    mx = fmaxf(mx, __shfl_xor(mx, off, 32));
  float sum = 0.0f;
  #pragma unroll
  for (int j = 0; j < 4; ++j) sum += __expf(vals[j] - mx);
  #pragma unroll
  for (int off = 16; off > 0; off >>= 1)
    sum += __shfl_xor(sum, off, 32);
  float lse = mx + __logf(sum);
  #pragma unroll
  for (int j = 0; j < 4; ++j) row[lane + 32 * j] = vals[j] - lse;
}

extern "C" void kernel_launch(void* const* d_in, const int* in_sizes, int n_in,
                              void* d_out, int out_size, void* d_ws, size_t ws_size,
                              hipStream_t stream) {
  const float* x    = (const float*)d_in[0];   // [50000,128]
  const int*   esrc = (const int*)  d_in[1];   // [800000]
  const int*   edst = (const int*)  d_in[2];   // [800000]
  const float* W1   = (const float*)d_in[3];   // [256,256]
  const float* b1   = (const float*)d_in[4];   // [256]
  const float* W2   = (const float*)d_in[5];   // [512,128]
  const float* b2   = (const float*)d_in[6];   // [128]
  float* out = (float*)d_out;                  // [50000,128]

  float* ws  = (float*)d_ws;
  float* deg = ws;                                   // 50000
  float* H0  = deg + NNODES;                         // 50000*256 (concat 1)
  float* H1  = H0 + (long long)NNODES * 256;         // 50000*256 (hidden)
  float* H2  = H1 + (long long)NNODES * 256;         // 50000*512 (concat 2)

  // zero deg + H0 (contiguous region)
  {
    long long z = NNODES + (long long)NNODES * 256;
    gs_zero_kernel<<<(unsigned)((z + 255) / 256), 256, 0, stream>>>(deg, z);
  }
  // degree
  gs_degree_kernel<<<(NEDGES + 255) / 256, 256, 0, stream>>>(edst, deg, NEDGES);
  // layer-1 neighbor sum into H0[:,128:256]
  gs_scatter_kernel<<<NEDGES / 8, 256, 0, stream>>>(x, 128, esrc, edst,
                                                    H0, 256, 128, NEDGES, 128);
  // H0[:,0:128]=x ; H0[:,128:256]*=inv_deg
  {
    long long t = (long long)NNODES * 128;
    gs_finalize_kernel<<<(unsigned)((t + 255) / 256), 256, 0, stream>>>(
        x, deg, H0, 128, 256, t);
  }
  // H1 = relu(H0 @ W1 + b1): 391 m-blocks x 4 col-groups, LDS = (256+4)*64*4 B
  {
    int mblocks = (NNODES / 16 + 7) / 8;             // 391
    size_t shmem = (size_t)(256 + 4) * 64 * sizeof(float);
    gs_gemm_lds_kernel<<<mblocks * 4, 256, shmem, stream>>>(
        H0, W1, b1, H1, NNODES, 256, 256, 1);
  }
  // zero H2
  {
    long long z = (long long)NNODES * 512;
    gs_zero_kernel<<<(unsigned)((z + 255) / 256), 256, 0, stream>>>(H2, z);
  }
  // layer-2 neighbor sum into H2[:,256:512]
  gs_scatter_kernel<<<NEDGES / 8, 256, 0, stream>>>(H1, 256, esrc, edst,
                                                    H2, 512, 256, NEDGES, 256);
  // H2[:,0:256]=H1 ; H2[:,256:512]*=inv_deg
  {
    long long t = (long long)NNODES * 256;
    gs_finalize_kernel<<<(unsigned)((t + 255) / 256), 256, 0, stream>>>(
        H1, deg, H2, 256, 512, t);
  }
  // out = H2 @ W2 + b2: 391 m-blocks x 2 col-groups, LDS = (512+4)*64*4 B
  {
    int mblocks = (NNODES / 16 + 7) / 8;             // 391
    size_t shmem = (size_t)(512 + 4) * 64 * sizeof(float);
    gs_gemm_lds_kernel<<<mblocks * 2, 256, shmem, stream>>>(
        H2, W2, b2, out, NNODES, 512, 128, 0);
  }
  // in-place log-softmax: 50000 waves -> 6250 blocks
  gs_logsoftmax_kernel<<<6250, 256, 0, stream>>>(out, NNODES);
}